// MultiClassSSIMLoss_11690900979898
// MI455X (gfx1250) — compile-verified
//
#include <hip/hip_runtime.h>

// ---- types for WMMA f32 16x16x4 ----
typedef float v2f __attribute__((ext_vector_type(2)));
typedef float v8f __attribute__((ext_vector_type(8)));

#define NCLS 16
#define HW   512
#define OUTW 502
#define TILE 32
#define IN_T 42     // TILE + WIN - 1
#define PSTR 43     // odd stride: bank-conflict avoidance
#define HSTR 33     // odd stride for h-buffers

// Gaussian window, size 11, sigma 1.5, normalized (matches _fspecial_gauss_1d)
__device__ __constant__ float GW[11] = {
    0.00102838f, 0.00759878f, 0.03600070f, 0.10936000f, 0.21300630f,
    0.26601110f, 0.21300630f, 0.10936000f, 0.03600070f, 0.00759878f,
    0.00102838f};

__device__ __forceinline__ float wband(int d) {
    return (d >= 0 && d <= 10) ? GW[d] : 0.0f;
}

__device__ __forceinline__ v8f wmma4(v2f a, v2f b, v8f c) {
    // D = A(16x4 f32) * B(4x16 f32) + C(16x16 f32)
    return __builtin_amdgcn_wmma_f32_16x16x4_f32(
        false, a, false, b, (short)0, c, false, false);
}

__global__ __launch_bounds__(256) void ssim_tile_kernel(
    const float* __restrict__ pred, const int* __restrict__ target,
    float* __restrict__ accum)
{
    __shared__ float sP[NCLS][IN_T][PSTR];     // softmax probs per channel
    __shared__ int   sT[IN_T][PSTR];           // target class ids
    __shared__ float sH[8][4][IN_T][HSTR];     // per-wave horiz-blur: p,p^2,t,p*t

    const int tx = blockIdx.x, ty = blockIdx.y, b = blockIdx.z;
    const int ix0 = tx * TILE, iy0 = ty * TILE;

    // ---------- cooperative softmax + staging ----------
    for (int idx = threadIdx.x; idx < IN_T * IN_T; idx += 256) {
        int r  = idx / IN_T;
        int cc = idx - r * IN_T;
        int gy = iy0 + r, gx = ix0 + cc;
        bool inb = (gy < HW) && (gx < HW);
        float v[NCLS];
        float m = -1e30f;
#pragma unroll
        for (int c = 0; c < NCLS; ++c) {
            float t = 0.0f;
            if (inb) t = pred[(((size_t)b * NCLS + c) * HW + gy) * HW + gx];
            v[c] = t;
            m = fmaxf(m, t);
        }
        float s = 0.0f;
#pragma unroll
        for (int c = 0; c < NCLS; ++c) { v[c] = __expf(v[c] - m); s += v[c]; }
        float inv = 1.0f / s;
#pragma unroll
        for (int c = 0; c < NCLS; ++c) sP[c][r][cc] = v[c] * inv;
        int tv = 255;
        if (inb) tv = target[((size_t)b * HW + gy) * HW + gx];
        sT[r][cc] = tv;
    }
    __syncthreads();

    const int lane = threadIdx.x & 31;
    const int w    = threadIdx.x >> 5;   // wave id 0..7
    const int half = lane >> 4;          // 0|1
    const int lm   = lane & 15;

    // Banded Gaussian fragments. Same per-lane values serve as the B matrix of
    // the horizontal pass (W[q][n] = w[q-n]) and the A matrix of the vertical
    // pass (W[m][q] = w[q-m]).
    v2f WF[7];
#pragma unroll
    for (int k = 0; k < 7; ++k) {
        int q0 = 4 * k + 2 * half;
        v2f t;
        t.x = wband(q0 - lm);
        t.y = wband(q0 + 1 - lm);
        WF[k] = t;
    }

    const float C1 = 1.0e-4f, C2 = 9.0e-4f;
    float part = 0.0f;

    for (int ci = 0; ci < 2; ++ci) {
        const int cls = w * 2 + ci;

        // -------- horizontal blur via WMMA: rows{0,16,26} x outcols{0,16} ----
        const int R0[3] = {0, 16, 26};
        for (int rg = 0; rg < 3; ++rg) {
            int r0 = R0[rg];
            for (int jg = 0; jg < 2; ++jg) {
                int j0 = jg * 16;
                v8f aP = {}, aPP = {}, aT = {}, aPT = {};
#pragma unroll
                for (int k = 0; k < 7; ++k) {
                    int q0 = 4 * k + 2 * half;
                    int c0 = j0 + q0;     if (c0 > 41) c0 = 41;  // zero-weight K cols
                    int c1 = j0 + q0 + 1; if (c1 > 41) c1 = 41;
                    int row = r0 + lm;
                    float p0 = sP[cls][row][c0];
                    float p1 = sP[cls][row][c1];
                    float t0 = (sT[row][c0] == cls) ? 1.0f : 0.0f;
                    float t1 = (sT[row][c1] == cls) ? 1.0f : 0.0f;
                    v2f A;
                    A.x = p0;      A.y = p1;      aP  = wmma4(A, WF[k], aP);
                    A.x = p0 * p0; A.y = p1 * p1; aPP = wmma4(A, WF[k], aPP);
                    A.x = t0;      A.y = t1;      aT  = wmma4(A, WF[k], aT);
                    A.x = p0 * t0; A.y = p1 * t1; aPT = wmma4(A, WF[k], aPT);
                }
#pragma unroll
                for (int v = 0; v < 8; ++v) {
                    int rr = r0 + v + 8 * half;   // D layout: M = v (+8 upper half)
                    int cw = j0 + lm;
                    sH[w][0][rr][cw] = aP[v];
                    sH[w][1][rr][cw] = aPP[v];
                    sH[w][2][rr][cw] = aT[v];
                    sH[w][3][rr][cw] = aPT[v];
                }
            }
        }

        // -------- vertical blur via WMMA + SSIM ----------
        for (int ig = 0; ig < 2; ++ig) {
            for (int jg = 0; jg < 2; ++jg) {
                int i0 = ig * 16, j0 = jg * 16;
                v8f mP = {}, mPP = {}, mT = {}, mPT = {};
#pragma unroll
                for (int k = 0; k < 7; ++k) {
                    int q0  = 4 * k + 2 * half;
                    int r0a = i0 + q0;     if (r0a > 41) r0a = 41; // zero-weight K rows
                    int r1a = i0 + q0 + 1; if (r1a > 41) r1a = 41;
                    int cw  = j0 + lm;
                    v2f B;
                    B.x = sH[w][0][r0a][cw]; B.y = sH[w][0][r1a][cw];
                    mP  = wmma4(WF[k], B, mP);
                    B.x = sH[w][1][r0a][cw]; B.y = sH[w][1][r1a][cw];
                    mPP = wmma4(WF[k], B, mPP);
                    B.x = sH[w][2][r0a][cw]; B.y = sH[w][2][r1a][cw];
                    mT  = wmma4(WF[k], B, mT);
                    B.x = sH[w][3][r0a][cw]; B.y = sH[w][3][r1a][cw];
                    mPT = wmma4(WF[k], B, mPT);
                }
#pragma unroll
                for (int v = 0; v < 8; ++v) {
                    float mu1 = mP[v], ep2 = mPP[v], mu2 = mT[v], ept = mPT[v];
                    float s1  = ep2 - mu1 * mu1;          // sigma1_sq
                    float s2  = mu2 - mu2 * mu2;          // sigma2_sq (t^2 == t)
                    float s12 = ept - mu1 * mu2;          // sigma12
                    float num = (2.0f * mu1 * mu2 + C1) * (2.0f * s12 + C2);
                    float den = (mu1 * mu1 + mu2 * mu2 + C1) * (s1 + s2 + C2);
                    int oy = iy0 + i0 + v + 8 * half;
                    int ox = ix0 + j0 + lm;
                    if (oy < OUTW && ox < OUTW) part += num / den;
                }
            }
        }
    }

    // wave32 reduction + one atomic per wave
#pragma unroll
    for (int off = 16; off > 0; off >>= 1) part += __shfl_xor(part, off, 32);
    if (lane == 0) atomicAdd(accum, part);
}

__global__ void ssim_zero_ws(float* ws) {
    if (threadIdx.x == 0) ws[0] = 0.0f;
}

__global__ void ssim_finalize(const float* ws, float* out) {
    if (threadIdx.x == 0) {
        const float invN = 1.0f / (16.0f * 8.0f * 502.0f * 502.0f);
        out[0] = 1.0f - ws[0] * invN;
    }
}

extern "C" void kernel_launch(void* const* d_in, const int* in_sizes, int n_in,
                              void* d_out, int out_size, void* d_ws, size_t ws_size,
                              hipStream_t stream) {
    (void)in_sizes; (void)n_in; (void)out_size; (void)ws_size;
    const float* pred   = (const float*)d_in[0];
    const int*   target = (const int*)d_in[1];
    float* ws  = (float*)d_ws;
    float* out = (float*)d_out;

    ssim_zero_ws<<<1, 32, 0, stream>>>(ws);
    dim3 grid(16, 16, 8);                 // 16x16 output tiles of 32, 8 batches
    ssim_tile_kernel<<<grid, 256, 0, stream>>>(pred, target, ws);
    ssim_finalize<<<1, 32, 0, stream>>>(ws, out);
}